// GCNConv_850403525191
// MI455X (gfx1250) — compile-verified
//
#include <hip/hip_runtime.h>
#include <hip/hip_bf16.h>

typedef __attribute__((ext_vector_type(2))) float v2f;
typedef __attribute__((ext_vector_type(8))) float v8f;

#define NFEAT 128

// ---------------- Kernel 1: in/out degrees via integer atomics ----------------
__global__ void k_degree(const int* __restrict__ src, const int* __restrict__ tgt,
                         int* __restrict__ dsrc, int* __restrict__ dtgt, int E) {
    int e = blockIdx.x * blockDim.x + threadIdx.x;
    if (e < E) {
        atomicAdd(&dsrc[src[e]], 1);
        atomicAdd(&dtgt[tgt[e]], 1);
    }
}

// ---------------- Kernel 2: rsqrt(max(deg,1)) scales ----------------
__global__ void k_scale(const int* __restrict__ dsrc, const int* __restrict__ dtgt,
                        float* __restrict__ ssrc, float* __restrict__ stgt, int N) {
    int i = blockIdx.x * blockDim.x + threadIdx.x;
    if (i < N) {
        int a = dsrc[i]; if (a < 1) a = 1;
        int b = dtgt[i]; if (b < 1) b = 1;
        ssrc[i] = rsqrtf((float)a);
        stgt[i] = rsqrtf((float)b);
    }
}

// ---------------- Kernel 3: edge gather + scatter-add (one wave32 per edge) ----------------
// pooled[t, :] += rsqrt(out_deg[s]) * x[s, :]   via native global f32 atomics.
__global__ void k_scatter(const float* __restrict__ x, const int* __restrict__ src,
                          const int* __restrict__ tgt, const float* __restrict__ ssrc,
                          float* __restrict__ pooled, int E) {
    int wid  = (int)((blockIdx.x * (unsigned)blockDim.x + threadIdx.x) >> 5);
    int lane = threadIdx.x & 31;
    if (wid >= E) return;
    int s = src[wid];
    int t = tgt[wid];
    float sc = ssrc[s];
    const float4* xr = (const float4*)(x + (size_t)s * NFEAT);
    float4 v = xr[lane];                       // 32 lanes x float4 = 128 feats, B128 coalesced
    float* pr = pooled + (size_t)t * NFEAT + lane * 4;
    unsafeAtomicAdd(pr + 0, sc * v.x);         // global_atomic_add_f32, no return
    unsafeAtomicAdd(pr + 1, sc * v.y);
    unsafeAtomicAdd(pr + 2, sc * v.z);
    unsafeAtomicAdd(pr + 3, sc * v.w);
}

// ---------------- Kernel 4: out = relu( (stgt[m]*pooled[m,:]) @ W + b ) ----------------
// One wave per 16-row strip; 8 accumulator tiles (16x16) cover all 128 output cols.
// Uses V_WMMA_F32_16X16X4_F32 (exact f32 math, matches the f32 reference GEMM).
__global__ void __launch_bounds__(256)
k_gemm_relu(const float* __restrict__ pooled, const float* __restrict__ stgt,
            const float* __restrict__ W, const float* __restrict__ bias,
            float* __restrict__ out, int M) {
    const int lane  = threadIdx.x & 31;
    const int wave  = threadIdx.x >> 5;
    const int strip = blockIdx.x * 8 + wave;   // wave-uniform -> EXEC all-1s inside
    if (strip * 16 >= M) return;

    const int l16   = lane & 15;
    const int khalf = lane >> 4;               // 0: K=k,k+1   1: K=k+2,k+3
    const int mrow  = strip * 16 + l16;

    const float  scale = stgt[mrow];
    const float2* arow = (const float2*)(pooled + (size_t)mrow * NFEAT);

    v8f acc[8];
#pragma unroll
    for (int t = 0; t < 8; ++t) acc[t] = (v8f){0.f,0.f,0.f,0.f,0.f,0.f,0.f,0.f};

    for (int kb = 0; kb < NFEAT; kb += 4) {
        const int k0 = kb + 2 * khalf;
        // A: 16x4 tile, row = l16, K = k0,k0+1 for this lane; fold in receiver scale
        float2 av = arow[k0 >> 1];
        v2f a; a.x = av.x * scale; a.y = av.y * scale;
        const float* w0 = W + (size_t)k0 * NFEAT + l16;   // B: rows K=k0,k0+1, col tile below
#pragma unroll
        for (int t = 0; t < 8; ++t) {
            v2f b;
            b.x = w0[t * 16];                  // W[k0    ][t*16 + l16]
            b.y = w0[NFEAT + t * 16];          // W[k0 + 1][t*16 + l16]
            acc[t] = __builtin_amdgcn_wmma_f32_16x16x4_f32(
                false, a, false, b, (short)0, acc[t], false, false);
        }
    }

    // C/D layout: VGPR i -> M=i (lanes 0-15), M=i+8 (lanes 16-31); N = l16 within tile.
#pragma unroll
    for (int t = 0; t < 8; ++t) {
        const int col = t * 16 + l16;
        const float bv = bias[col];
#pragma unroll
        for (int i = 0; i < 8; ++i) {
            const int row = strip * 16 + khalf * 8 + i;
            float v = acc[t][i] + bv;
            out[(size_t)row * NFEAT + col] = v > 0.f ? v : 0.f;
        }
    }
}

extern "C" void kernel_launch(void* const* d_in, const int* in_sizes, int n_in,
                              void* d_out, int out_size, void* d_ws, size_t ws_size,
                              hipStream_t stream) {
    const float* x    = (const float*)d_in[0];
    const int*   src  = (const int*)d_in[1];
    const int*   tgt  = (const int*)d_in[2];
    const float* W    = (const float*)d_in[3];
    const float* bias = (const float*)d_in[4];
    float*       out  = (float*)d_out;

    const int N = in_sizes[0] / NFEAT;   // 50000 nodes
    const int E = in_sizes[1];           // 800000 edges

    // Workspace layout: [deg_src:int N][deg_tgt:int N][ssrc:f32 N][stgt:f32 N][pooled:f32 N*128]
    int*   dsrc   = (int*)d_ws;
    int*   dtgt   = dsrc + N;
    float* ssrc   = (float*)(dtgt + N);
    float* stgt   = ssrc + N;
    float* pooled = stgt + N;

    hipMemsetAsync(dsrc,   0, sizeof(int)   * 2 * (size_t)N,       stream);
    hipMemsetAsync(pooled, 0, sizeof(float) * (size_t)N * NFEAT,   stream);

    k_degree <<<(E + 255) / 256, 256, 0, stream>>>(src, tgt, dsrc, dtgt, E);
    k_scale  <<<(N + 255) / 256, 256, 0, stream>>>(dsrc, dtgt, ssrc, stgt, N);
    k_scatter<<<(E + 7) / 8,     256, 0, stream>>>(x, src, tgt, ssrc, pooled, E);

    const int strips = (N + 15) / 16;    // 3125
    k_gemm_relu<<<(strips + 7) / 8, 256, 0, stream>>>(pooled, stgt, W, bias, out, N);
}